// TwoGCN_19662360281499
// MI455X (gfx1250) — compile-verified
//
#include <hip/hip_runtime.h>

typedef float v2f __attribute__((ext_vector_type(2)));
typedef float v8f __attribute__((ext_vector_type(8)));

#define CH 64  // hidden/output channel count

__device__ __forceinline__ void atomAddF(float* p, float v) {
    // relaxed, device-scope, no return -> global_atomic_add_f32 (STOREcnt path)
    __hip_atomic_fetch_add(p, v, __ATOMIC_RELAXED, __HIP_MEMORY_SCOPE_AGENT);
}

// ---------------- degree / norm ----------------

__global__ void init_deg_k(float* __restrict__ deg, int N) {
    int i = blockIdx.x * blockDim.x + threadIdx.x;
    if (i < N) deg[i] = 1.0f;  // self-loop
}

__global__ void count_deg_k(const int* __restrict__ dst, float* __restrict__ deg, int E) {
    int e = blockIdx.x * blockDim.x + threadIdx.x;
    if (e < E) atomAddF(&deg[dst[e]], 1.0f);
}

__global__ void rsqrt_k(float* __restrict__ d, int N) {
    int i = blockIdx.x * blockDim.x + threadIdx.x;
    if (i < N) d[i] = rsqrtf(d[i]);  // deg >= 1 always (self-loops)
}

// ---------------- GEMM: Y = X * W^T via V_WMMA_F32_16X16X4_F32 ----------------
// X: [N, K] row-major, W: [CH, K] row-major (torch Linear layout), Y: [N, CH]
// One wave computes one 16x16 tile of Y. K-loop in steps of 4.
// A (16x4 f32) lane layout: lane l holds A[l%16][2*(l/16) + {0,1}] in v[0],v[1]
// B (4x16 f32) lane layout: lane l holds B[2*(l/16) + {0,1}][l%16] = W[n0+l%16][k0+2*(l/16)+{0,1}]
// D (16x16 f32): VGPR r, lane l -> Y[m0 + r + 8*(l/16)][n0 + l%16]
__global__ void gemm_xwT_k(const float* __restrict__ X, const float* __restrict__ W,
                           float* __restrict__ Y, int N, int K) {
    const int lane = threadIdx.x & 31;
    const int wave = (blockIdx.x * blockDim.x + threadIdx.x) >> 5;
    const int tilesN = CH / 16;                 // 4 tiles across the 64 output cols
    const int band   = wave / tilesN;           // 16-row band of Y
    const int tn     = wave % tilesN;
    const int m0 = band * 16;
    const int n0 = tn * 16;
    if (m0 >= N) return;                        // wave-uniform guard (EXEC stays full)

    const int l16   = lane & 15;
    const int khalf = (lane >> 4) * 2;

    const float* xrow = X + (size_t)(m0 + l16) * K;
    const float* wrow = W + (size_t)(n0 + l16) * K;

    v8f c = {0.f, 0.f, 0.f, 0.f, 0.f, 0.f, 0.f, 0.f};
    for (int k0 = 0; k0 < K; k0 += 4) {
        v2f a = *(const v2f*)(xrow + k0 + khalf);
        v2f b = *(const v2f*)(wrow + k0 + khalf);
        c = __builtin_amdgcn_wmma_f32_16x16x4_f32(
                /*neg_a=*/false, a, /*neg_b=*/false, b,
                /*c_mod=*/(short)0, c, /*reuse_a=*/false, /*reuse_b=*/false);
    }

    const int mbase = m0 + (lane >> 4) * 8;
    const int ncol  = n0 + l16;
#pragma unroll
    for (int r = 0; r < 8; ++r) {
        Y[(size_t)(mbase + r) * CH + ncol] = c[r];
    }
}

// ---------------- aggregation ----------------

// agg[i][c] = y[i][c] * dinv[i]^2 (self-loop) + bias[c]   (float4-vectorized)
__global__ void init_agg4_k(const float4* __restrict__ Y4, const float* __restrict__ dinv,
                            const float4* __restrict__ bias4, float4* __restrict__ agg4,
                            int n4) {   // n4 = N * CH / 4
    int t = blockIdx.x * blockDim.x + threadIdx.x;
    if (t >= n4) return;
    int i  = t >> 4;        // node index: 16 float4 per 64-ch row
    int c4 = t & 15;        // float4 index within row
    float di = dinv[i];
    float s  = di * di;
    float4 y = Y4[t];
    float4 b = bias4[c4];
    float4 r;
    r.x = y.x * s + b.x;
    r.y = y.y * s + b.y;
    r.z = y.z * s + b.z;
    r.w = y.w * s + b.w;
    agg4[t] = r;
}

// one wave32 per edge; lane handles channels {2*lane, 2*lane+1} (coalesced 256B row)
__global__ void scatter_k(const float* __restrict__ Y, const int* __restrict__ src,
                          const int* __restrict__ dst, const float* __restrict__ dinv,
                          float* __restrict__ agg, int E) {
    int e    = (blockIdx.x * blockDim.x + threadIdx.x) >> 5;
    int lane = threadIdx.x & 31;
    if (e >= E) return;
    int s = src[e];
    int d = dst[e];
    float nrm = dinv[s] * dinv[d];
    const float2* yrow = (const float2*)(Y + (size_t)s * CH);
    float* arow = agg + (size_t)d * CH;
    float2 v = yrow[lane];
    atomAddF(arow + 2 * lane,     v.x * nrm);
    atomAddF(arow + 2 * lane + 1, v.y * nrm);
}

__global__ void relu4_k(const float4* __restrict__ in, float4* __restrict__ out, int n4) {
    int t = blockIdx.x * blockDim.x + threadIdx.x;
    if (t >= n4) return;
    float4 v = in[t];
    v.x = fmaxf(v.x, 0.0f);
    v.y = fmaxf(v.y, 0.0f);
    v.z = fmaxf(v.z, 0.0f);
    v.w = fmaxf(v.w, 0.0f);
    out[t] = v;
}

// ---------------- driver ----------------

extern "C" void kernel_launch(void* const* d_in, const int* in_sizes, int n_in,
                              void* d_out, int out_size, void* d_ws, size_t ws_size,
                              hipStream_t stream) {
    const float* x    = (const float*)d_in[0];
    const int*   edge = (const int*)  d_in[1];
    const float* W1   = (const float*)d_in[2];
    const float* b1   = (const float*)d_in[3];
    const float* W2   = (const float*)d_in[4];
    const float* b2   = (const float*)d_in[5];

    const int N = in_sizes[0] / 256;   // 100000
    const int E = in_sizes[1] / 2;     // 1200000
    const int* src = edge;
    const int* dst = edge + E;

    // workspace: dinv[N] | y[N*64] | agg[N*64]
    float* dinv = (float*)d_ws;
    float* y    = dinv + N;
    float* agg  = y + (size_t)N * CH;

    const int n4 = N * CH / 4;
    const int bT = 256;

    // degree -> dinv
    init_deg_k <<<(N + bT - 1) / bT, bT, 0, stream>>>(dinv, N);
    count_deg_k<<<(E + bT - 1) / bT, bT, 0, stream>>>(dst, dinv, E);
    rsqrt_k    <<<(N + bT - 1) / bT, bT, 0, stream>>>(dinv, N);

    // GEMM grid: one wave per 16x16 tile; (N/16)*4 tiles, 8 waves per 256-thread block
    const int tiles      = (N / 16) * (CH / 16);
    const int gemmBlocks = (tiles * 32 + bT - 1) / bT;

    // ---- layer 1 ----
    gemm_xwT_k <<<gemmBlocks, bT, 0, stream>>>(x, W1, y, N, 256);
    init_agg4_k<<<(n4 + bT - 1) / bT, bT, 0, stream>>>(
        (const float4*)y, dinv, (const float4*)b1, (float4*)agg, n4);
    scatter_k  <<<(E + 7) / 8, bT, 0, stream>>>(y, src, dst, dinv, agg, E);
    relu4_k    <<<(n4 + bT - 1) / bT, bT, 0, stream>>>(
        (const float4*)agg, (float4*)agg, n4);                         // h in-place

    // ---- layer 2 ----
    gemm_xwT_k <<<gemmBlocks, bT, 0, stream>>>(agg, W2, y, N, CH);
    init_agg4_k<<<(n4 + bT - 1) / bT, bT, 0, stream>>>(
        (const float4*)y, dinv, (const float4*)b2, (float4*)agg, n4);
    scatter_k  <<<(E + 7) / 8, bT, 0, stream>>>(y, src, dst, dinv, agg, E);
    relu4_k    <<<(n4 + bT - 1) / bT, bT, 0, stream>>>(
        (const float4*)agg, (float4*)d_out, n4);
}